// GNN_Backbone_4776003633767
// MI455X (gfx1250) — compile-verified
//
#include <hip/hip_runtime.h>

// ---------------------------------------------------------------------------
// GraphSAGE 2-layer backbone for MI455X (gfx1250, wave32, WMMA)
//   N nodes, E edges, F: 64 -> 128 -> 128
//   Bottleneck: edge scatter (L2-resident atomics). GEMMs use f32 WMMA
//   (16x16x4) to keep full fp32 precision -- GEMM cost is negligible.
// ---------------------------------------------------------------------------

typedef float v2f __attribute__((ext_vector_type(2)));
typedef float v8f __attribute__((ext_vector_type(8)));

#define F_HID 128
#define F_OUT 128

// ---------------------------------------------------------------------------
// Scatter: for each edge e, agg[dst[e]] += feat[src[e]] (F floats).
// One thread per (edge, 4-float chunk). Optionally accumulates degree.
// ---------------------------------------------------------------------------
template <int F>
__global__ __launch_bounds__(256) void sage_scatter_kernel(
    const float* __restrict__ feat,   // [N, F]
    const int*   __restrict__ src,    // [E]
    const int*   __restrict__ dst,    // [E]
    float*       __restrict__ agg,    // [N, F]  (pre-zeroed)
    float*       __restrict__ deg,    // [N] or nullptr (pre-zeroed)
    int E)
{
    constexpr int CHUNKS = F / 4;
    int tid = blockIdx.x * 256 + threadIdx.x;
    int e   = tid / CHUNKS;
    if (e >= E) return;
    int ch  = (tid - e * CHUNKS) * 4;

    int s = src[e];
    int d = dst[e];

    const float4 v = *(const float4*)(feat + (size_t)s * F + ch);
    float* dp = agg + (size_t)d * F + ch;
    atomicAdd(dp + 0, v.x);
    atomicAdd(dp + 1, v.y);
    atomicAdd(dp + 2, v.z);
    atomicAdd(dp + 3, v.w);
    if (deg != nullptr && ch == 0) {
        atomicAdd(deg + d, 1.0f);
    }
}

// deg -> 1 / max(deg, 1), in place
__global__ __launch_bounds__(256) void deg_inv_kernel(float* __restrict__ deg, int N)
{
    int i = blockIdx.x * 256 + threadIdx.x;
    if (i < N) deg[i] = 1.0f / fmaxf(deg[i], 1.0f);
}

// ---------------------------------------------------------------------------
// Fused SAGE GEMM:
//   out = [relu]( (agg * dinv[row]) @ Wl^T + bias + xin @ Wr^T )
// Block = 256 threads = 8 waves. Block handles a 16-row node tile; wave w
// owns output columns [16w, 16w+16). Each wave runs two K-loops of
// V_WMMA_F32_16X16X4_F32 into one f32 accumulator.
//
// Fragment layout per CDNA5 ISA 7.12.2 (32-bit, wave32):
//   A 16x4 : lane m in [0,16) holds A[m][k0], A[m][k0+1] in v[0],v[1];
//            lanes 16..31 hold the same rows at k0+2, k0+3.
//   B 4x16 : lane n holds B[k0][n], B[k0+1][n]; upper half-wave k0+2/k0+3.
//   C 16x16: vgpr i, lanes 0-15 -> (M=i, N=lane); lanes 16-31 -> (M=i+8).
// Both A and B fragments are contiguous-in-K -> single float2 loads.
// ---------------------------------------------------------------------------
template <int K, bool RELU>
__global__ __launch_bounds__(256) void sage_gemm_kernel(
    const float* __restrict__ agg,    // [N, K]
    const float* __restrict__ xin,    // [N, K]
    const float* __restrict__ Wl,     // [128, K] row-major
    const float* __restrict__ Wr,     // [128, K] row-major
    const float* __restrict__ bias,   // [128]
    const float* __restrict__ dinv,   // [N]
    float*       __restrict__ out,    // [N, 128]
    int N)
{
    const int lane = threadIdx.x & 31;
    const int wave = threadIdx.x >> 5;
    const int m    = lane & 15;            // A row / B col within tile
    const int koff = (lane >> 4) * 2;      // upper half-wave: K offset +2

    int rowA = blockIdx.x * 16 + m;
    if (rowA > N - 1) rowA = N - 1;        // clamp loads; EXEC stays all-ones
    const int col = wave * 16 + m;

    const float dscale = dinv[rowA];
    const float* aggRow = agg + (size_t)rowA * K + koff;
    const float* xRow   = xin + (size_t)rowA * K + koff;
    const float* WlRow  = Wl  + (size_t)col  * K + koff;
    const float* WrRow  = Wr  + (size_t)col  * K + koff;

    v8f c = {};
#pragma unroll
    for (int k = 0; k < K; k += 4) {
        v2f a = *(const v2f*)(aggRow + k);
        a.x *= dscale;
        a.y *= dscale;
        v2f b = *(const v2f*)(WlRow + k);
        c = __builtin_amdgcn_wmma_f32_16x16x4_f32(
                /*neg_a=*/false, a, /*neg_b=*/false, b,
                /*c_mod=*/(short)0, c, /*reuse_a=*/false, /*reuse_b=*/false);
    }
#pragma unroll
    for (int k = 0; k < K; k += 4) {
        v2f a = *(const v2f*)(xRow + k);
        v2f b = *(const v2f*)(WrRow + k);
        c = __builtin_amdgcn_wmma_f32_16x16x4_f32(
                false, a, false, b, (short)0, c, false, false);
    }

    // Store C tile: vgpr i -> row (base + i), upper half-wave rows +8.
    const int outCol  = wave * 16 + (lane & 15);
    const int rowBase = blockIdx.x * 16 + ((lane >> 4) * 8);
    const float bv = bias[outCol];
#pragma unroll
    for (int i = 0; i < 8; ++i) {
        int r = rowBase + i;
        if (r < N) {
            float v = c[i] + bv;
            if (RELU) v = fmaxf(v, 0.0f);
            out[(size_t)r * 128 + outCol] = v;
        }
    }
}

// ---------------------------------------------------------------------------
// Host-side orchestration (all ops on `stream`, graph-capturable)
// ---------------------------------------------------------------------------
extern "C" void kernel_launch(void* const* d_in, const int* in_sizes, int n_in,
                              void* d_out, int out_size, void* d_ws, size_t ws_size,
                              hipStream_t stream)
{
    // Inputs per setup_inputs() order:
    const float* x    = (const float*)d_in[0];   // [N, 64]
    const int*   ei   = (const int*)  d_in[1];   // [2, E]
    const float* W1l  = (const float*)d_in[2];   // [128, 64]
    const float* b1   = (const float*)d_in[3];   // [128]
    const float* W1r  = (const float*)d_in[4];   // [128, 64]
    const float* W2l  = (const float*)d_in[5];   // [128, 128]
    const float* b2   = (const float*)d_in[6];   // [128]
    const float* W2r  = (const float*)d_in[7];   // [128, 128]
    float*       out  = (float*)d_out;           // [N, 128]

    const int N = in_sizes[0] / 64;
    const int E = in_sizes[1] / 2;
    const int* src = ei;
    const int* dst = ei + E;

    // Workspace layout (floats): deg/deg_inv [N] | agg [N*128] | h [N*128]
    float* deg = (float*)d_ws;
    size_t nAlign = ((size_t)N + 3) & ~(size_t)3;   // keep agg 16B-aligned
    float* aggbuf = deg + nAlign;
    float* h      = aggbuf + (size_t)N * 128;

    // ---- Layer 1 -----------------------------------------------------------
    hipMemsetAsync(deg,    0, (size_t)N * sizeof(float), stream);
    hipMemsetAsync(aggbuf, 0, (size_t)N * 64 * sizeof(float), stream);

    {
        int threads = E * (64 / 4);
        int blocks  = (threads + 255) / 256;
        sage_scatter_kernel<64><<<blocks, 256, 0, stream>>>(
            x, src, dst, aggbuf, deg, E);
    }
    deg_inv_kernel<<<(N + 255) / 256, 256, 0, stream>>>(deg, N);

    sage_gemm_kernel<64, true><<<(N + 15) / 16, 256, 0, stream>>>(
        aggbuf, x, W1l, W1r, b1, deg, h, N);

    // ---- Layer 2 -----------------------------------------------------------
    hipMemsetAsync(aggbuf, 0, (size_t)N * 128 * sizeof(float), stream);
    {
        int threads = E * (128 / 4);
        int blocks  = (threads + 255) / 256;
        sage_scatter_kernel<128><<<blocks, 256, 0, stream>>>(
            h, src, dst, aggbuf, nullptr, E);
    }
    sage_gemm_kernel<128, false><<<(N + 15) / 16, 256, 0, stream>>>(
        aggbuf, h, W2l, W2r, b2, deg, out, N);
}